// GraphSAGE_AML_32246614458737
// MI455X (gfx1250) — compile-verified
//
#include <hip/hip_runtime.h>
#include <hip/hip_bf16.h>

typedef float v2f __attribute__((ext_vector_type(2)));
typedef float v8f __attribute__((ext_vector_type(8)));
typedef int   v4i __attribute__((vector_size(16)));

#define EPS 1e-5f

// Detect gfx1250 async global->LDS builtins (ASYNCcnt-tracked DMA).
#if defined(__has_builtin)
#if __has_builtin(__builtin_amdgcn_global_load_async_to_lds_b128)
#define HAVE_ASYNC_LDS 1
#endif
#if __has_builtin(__builtin_amdgcn_s_wait_asynccnt)
#define HAVE_WAIT_ASYNC 1
#endif
#endif

__device__ __forceinline__ void wait_asynccnt0() {
#if defined(HAVE_WAIT_ASYNC)
    __builtin_amdgcn_s_wait_asynccnt(0);
#elif defined(HAVE_ASYNC_LDS)
    asm volatile("s_wait_asynccnt 0" ::: "memory");
#endif
}

#if defined(HAVE_ASYNC_LDS)
// Async DMA of one 16-byte granule: global (AS1) -> LDS (AS3).
__device__ __forceinline__ void async_copy16(const float* gsrc, float* ldst) {
    v4i* g_generic = (v4i*)gsrc;   // C-style cast: drops const, retypes pointee
    v4i* l_generic = (v4i*)ldst;
    __builtin_amdgcn_global_load_async_to_lds_b128(
        (__attribute__((address_space(1))) v4i*)g_generic,
        (__attribute__((address_space(3))) v4i*)l_generic, 0, 0);
}
#endif

// ---------------------------------------------------------------------------
// Utility kernels
// ---------------------------------------------------------------------------
__global__ void zero_kernel(float* __restrict__ p, int n) {
    int i = blockIdx.x * blockDim.x + threadIdx.x;
    if (i < n) p[i] = 0.0f;
}

__global__ void deg_kernel(const int* __restrict__ dst, float* __restrict__ deg, int E) {
    int e = blockIdx.x * blockDim.x + threadIdx.x;
    if (e < E) atomicAdd(deg + dst[e], 1.0f);
}

__global__ void inv_kernel(float* __restrict__ deg, int n) {
    int i = blockIdx.x * blockDim.x + threadIdx.x;
    if (i < n) deg[i] = 1.0f / fmaxf(deg[i], 1.0f);
}

// ---------------------------------------------------------------------------
// Edge scatter: agg[dst] += x[src]  (float4 gather, f32 atomics into L2-resident agg)
// ---------------------------------------------------------------------------
__global__ __launch_bounds__(256)
void scatter_kernel(const float* __restrict__ x, const int* __restrict__ src,
                    const int* __restrict__ dst, float* __restrict__ agg,
                    int E, int log2chunks, int din) {
    long long idx = (long long)blockIdx.x * blockDim.x + threadIdx.x;
    long long total = (long long)E << log2chunks;
    if (idx >= total) return;
    int e = (int)(idx >> log2chunks);
    int c = (int)(idx & ((1 << log2chunks) - 1));
    int s = src[e], d = dst[e];
    const float4 v = *reinterpret_cast<const float4*>(x + (size_t)s * din + c * 4);
    float* base = agg + (size_t)d * din + c * 4;
    atomicAdd(base + 0, v.x);
    atomicAdd(base + 1, v.y);
    atomicAdd(base + 2, v.z);
    atomicAdd(base + 3, v.w);
}

// ---------------------------------------------------------------------------
// Fused SAGE layer: out = relu(BN(agg*invdeg @ Wl^T + x @ Wr^T + b)) [+ x]
// One wave -> TWO 16x16 output tiles sharing every B fetch (half the DS
// traffic, 2 WMMAs per operand pair). 8 waves/block -> 64 nodes x 64 cols.
// Weights staged in LDS via async global->LDS DMA when available.
// ---------------------------------------------------------------------------
template <int DIN, bool RES>
__global__ __launch_bounds__(256)
void sage_layer_kernel(const float* __restrict__ xin,
                       const float* __restrict__ agg,
                       const float* __restrict__ invdeg,
                       const float* __restrict__ Wl,
                       const float* __restrict__ Wr,
                       const float* __restrict__ bias,
                       const float* __restrict__ gamma,
                       const float* __restrict__ beta,
                       const float* __restrict__ rmean,
                       const float* __restrict__ rvar,
                       float* __restrict__ out,
                       int nNodes) {
    constexpr int H = 64;
    __shared__ float sWl[H * DIN];
    __shared__ float sWr[H * DIN];

    const int tid = threadIdx.x;
#if defined(HAVE_ASYNC_LDS)
    {
        constexpr int vec = (H * DIN) / 4;   // float4 granules (512 or 1024: /256 exact)
        for (int i = tid; i < vec; i += 256) {
            async_copy16(Wl + i * 4, sWl + i * 4);
            async_copy16(Wr + i * 4, sWr + i * 4);
        }
        wait_asynccnt0();
    }
#else
    for (int i = tid; i < H * DIN; i += 256) {
        sWl[i] = Wl[i];
        sWr[i] = Wr[i];
    }
#endif
    __syncthreads();

    const int lane = tid & 31;
    const int wave = tid >> 5;
    const int ln15 = lane & 15;
    const int hi   = lane >> 4;          // 0: lanes 0-15, 1: lanes 16-31

    const int hBase    = (wave & 3) * 16;
    const int nodeBase = blockIdx.x * 64 + (wave >> 2) * 32;   // two 16-row tiles

    // A rows for this lane (ISA 7.12.2: 32-bit A 16x4, lanes>=16 hold K+2)
    int row0 = nodeBase + ln15;
    int row1 = nodeBase + 16 + ln15;
    if (row0 >= nNodes) row0 = nNodes - 1;     // clamp loads, keep EXEC full
    if (row1 >= nNodes) row1 = nNodes - 1;
    const float invd0 = invdeg[row0];
    const float invd1 = invdeg[row1];

    const int colB = hBase + ln15;             // B column for this lane

    const float* aggRow0 = agg + (size_t)row0 * DIN;
    const float* aggRow1 = agg + (size_t)row1 * DIN;
    const float* xRow0   = xin + (size_t)row0 * DIN;
    const float* xRow1   = xin + (size_t)row1 * DIN;
    const float* wlCol   = sWl + colB * DIN;
    const float* wrCol   = sWr + colB * DIN;

    v8f c0 = {0.f, 0.f, 0.f, 0.f, 0.f, 0.f, 0.f, 0.f};
    v8f c1 = {0.f, 0.f, 0.f, 0.f, 0.f, 0.f, 0.f, 0.f};

    // First half of K: mean-aggregated neighbors vs W_l
#pragma unroll
    for (int kk = 0; kk < DIN / 4; ++kk) {
        const int kOff = kk * 4 + hi * 2;      // j-th VGPR adds +j
        v2f b = *reinterpret_cast<const v2f*>(wlCol + kOff);   // ds_load_b64, shared
        v2f a0 = *reinterpret_cast<const v2f*>(aggRow0 + kOff);
        v2f a1 = *reinterpret_cast<const v2f*>(aggRow1 + kOff);
        a0[0] *= invd0; a0[1] *= invd0;
        a1[0] *= invd1; a1[1] *= invd1;
        c0 = __builtin_amdgcn_wmma_f32_16x16x4_f32(false, a0, false, b,
                                                   (short)0, c0, false, false);
        c1 = __builtin_amdgcn_wmma_f32_16x16x4_f32(false, a1, false, b,
                                                   (short)0, c1, false, false);
    }
    // Second half of K: self features vs W_r
#pragma unroll
    for (int kk = 0; kk < DIN / 4; ++kk) {
        const int kOff = kk * 4 + hi * 2;
        v2f b = *reinterpret_cast<const v2f*>(wrCol + kOff);
        v2f a0 = *reinterpret_cast<const v2f*>(xRow0 + kOff);
        v2f a1 = *reinterpret_cast<const v2f*>(xRow1 + kOff);
        c0 = __builtin_amdgcn_wmma_f32_16x16x4_f32(false, a0, false, b,
                                                   (short)0, c0, false, false);
        c1 = __builtin_amdgcn_wmma_f32_16x16x4_f32(false, a1, false, b,
                                                   (short)0, c1, false, false);
    }

    // Epilogue. C/D layout: lane ln15 = column, VGPR r -> row r + 8*hi.
    // Column (hidden index) is constant per lane -> scalar BN fold.
    const int h = hBase + ln15;
    const float scale = gamma[h] * rsqrtf(rvar[h] + EPS);
    const float shift = beta[h] + (bias[h] - rmean[h]) * scale;

#pragma unroll
    for (int r = 0; r < 8; ++r) {
        const int m = r + hi * 8;
        int node = nodeBase + m;
        if (node < nNodes) {
            float v = fmaxf(c0[r] * scale + shift, 0.0f);
            if (RES) v += xin[(size_t)node * H + h];
            out[(size_t)node * H + h] = v;
        }
        node = nodeBase + 16 + m;
        if (node < nNodes) {
            float v = fmaxf(c1[r] * scale + shift, 0.0f);
            if (RES) v += xin[(size_t)node * H + h];
            out[(size_t)node * H + h] = v;
        }
    }
}

// ---------------------------------------------------------------------------
// Classifier: logits = x @ W_cls^T + b_cls ; log_softmax over C=2
// ---------------------------------------------------------------------------
__global__ __launch_bounds__(256)
void cls_kernel(const float* __restrict__ x, const float* __restrict__ Wc,
                const float* __restrict__ bc, float* __restrict__ out, int nNodes) {
    int i = blockIdx.x * blockDim.x + threadIdx.x;
    if (i >= nNodes) return;
    const float* row = x + (size_t)i * 64;
    float l0 = bc[0], l1 = bc[1];
#pragma unroll
    for (int k = 0; k < 64; k += 4) {
        const float4 v  = *reinterpret_cast<const float4*>(row + k);
        const float4 w0 = *reinterpret_cast<const float4*>(Wc + k);
        const float4 w1 = *reinterpret_cast<const float4*>(Wc + 64 + k);
        l0 += v.x * w0.x + v.y * w0.y + v.z * w0.z + v.w * w0.w;
        l1 += v.x * w1.x + v.y * w1.y + v.z * w1.z + v.w * w1.w;
    }
    const float m = fmaxf(l0, l1);
    const float lse = m + logf(expf(l0 - m) + expf(l1 - m));
    out[2 * i + 0] = l0 - lse;
    out[2 * i + 1] = l1 - lse;
}

// ---------------------------------------------------------------------------
// Launch
// ---------------------------------------------------------------------------
extern "C" void kernel_launch(void* const* d_in, const int* in_sizes, int n_in,
                              void* d_out, int out_size, void* d_ws, size_t ws_size,
                              hipStream_t stream) {
    const int DIN = 32, H = 64;
    const float* x  = (const float*)d_in[0];
    const int*   ei = (const int*)d_in[1];
    const int N = in_sizes[0] / DIN;
    const int E = in_sizes[1] / 2;
    const int* src = ei;
    const int* dst = ei + E;

    // per-layer params start at index 2: W_l, b, W_r, g, be, rm, rv
    const float* p[3][7];
    for (int l = 0; l < 3; ++l)
        for (int j = 0; j < 7; ++j)
            p[l][j] = (const float*)d_in[2 + 7 * l + j];
    const float* Wcls = (const float*)d_in[23];
    const float* bcls = (const float*)d_in[24];

    float* ws   = (float*)d_ws;
    float* deg  = ws;                       // N  (becomes inverse degree)
    float* agg  = deg + N;                  // N*H
    float* bufA = agg + (size_t)N * H;      // N*H
    float* bufB = bufA + (size_t)N * H;     // N*H
    float* out  = (float*)d_out;

    const int T = 256;
    const int gN    = (N + T - 1) / T;
    const int gE    = (E + T - 1) / T;
    const int gNH   = (N * H + T - 1) / T;
    const int gN32  = (N * 32 + T - 1) / T;
    const int gTile = (N + 63) / 64;

    // Degrees -> inverse degrees (edges don't change across layers)
    zero_kernel<<<gN, T, 0, stream>>>(deg, N);
    deg_kernel<<<gE, T, 0, stream>>>(dst, deg, E);
    inv_kernel<<<gN, T, 0, stream>>>(deg, N);

    // ---- Layer 0: DIN=32, no residual ----
    zero_kernel<<<gN32, T, 0, stream>>>(agg, N * 32);
    {
        long long total = (long long)E << 3;          // 32/4 = 8 chunks
        int g = (int)((total + T - 1) / T);
        scatter_kernel<<<g, T, 0, stream>>>(x, src, dst, agg, E, 3, 32);
    }
    sage_layer_kernel<32, false><<<gTile, T, 0, stream>>>(
        x, agg, deg, p[0][0], p[0][2], p[0][1], p[0][3], p[0][4], p[0][5], p[0][6],
        bufA, N);

    // ---- Layer 1: DIN=64, residual ----
    zero_kernel<<<gNH, T, 0, stream>>>(agg, N * H);
    {
        long long total = (long long)E << 4;          // 64/4 = 16 chunks
        int g = (int)((total + T - 1) / T);
        scatter_kernel<<<g, T, 0, stream>>>(bufA, src, dst, agg, E, 4, 64);
    }
    sage_layer_kernel<64, true><<<gTile, T, 0, stream>>>(
        bufA, agg, deg, p[1][0], p[1][2], p[1][1], p[1][3], p[1][4], p[1][5], p[1][6],
        bufB, N);

    // ---- Layer 2: DIN=64, residual ----
    zero_kernel<<<gNH, T, 0, stream>>>(agg, N * H);
    {
        long long total = (long long)E << 4;
        int g = (int)((total + T - 1) / T);
        scatter_kernel<<<g, T, 0, stream>>>(bufB, src, dst, agg, E, 4, 64);
    }
    sage_layer_kernel<64, true><<<gTile, T, 0, stream>>>(
        bufB, agg, deg, p[2][0], p[2][2], p[2][1], p[2][3], p[2][4], p[2][5], p[2][6],
        bufA, N);

    // ---- Classifier + log-softmax ----
    cls_kernel<<<gN, T, 0, stream>>>(bufA, Wcls, bcls, out, N);
}